// QueryAndGroup_77343771066371
// MI455X (gfx1250) — compile-verified
//
#include <hip/hip_runtime.h>

typedef float v2f __attribute__((ext_vector_type(2)));
typedef float v8f __attribute__((ext_vector_type(8)));

#define BATCH   8
#define NPTS    16384
#define NQ      2048
#define NS      32
#define NC      64
#define COUT    67          // 3 xyz channels + 64 feature channels
#define WPB     4           // waves per block in KNN kernel
#define SROW    20          // padded score-row stride (dwords): conflict-free b128 reads
#define NTILES  (NPTS / 16)

// ---------------------------------------------------------------------------
// Kernel 0: pack per-point WMMA B-operand: ptp[b][n] = (-2x, -2y, -2z, |p|^2)
// 2 MB total -> L2 resident, re-read by every query tile of the batch.
// ---------------------------------------------------------------------------
__global__ __launch_bounds__(256) void pack_points_kernel(
    const float* __restrict__ xyz, float4* __restrict__ ptp)
{
    int i = blockIdx.x * blockDim.x + threadIdx.x;       // over BATCH*NPTS
    if (i >= BATCH * NPTS) return;
    float x = xyz[(size_t)i * 3 + 0];
    float y = xyz[(size_t)i * 3 + 1];
    float z = xyz[(size_t)i * 3 + 2];
    ptp[i] = make_float4(-2.0f * x, -2.0f * y, -2.0f * z, x * x + y * y + z * z);
}

// ---------------------------------------------------------------------------
// Kernel 1: fused WMMA distance + top-32 selection.
// One wave32 owns a 16-query tile. Per 16-point tile:
//   score[16x16] = A(16x4) * B(4x16) via V_WMMA_F32_16X16X4_F32
//   A row m = [qx, qy, qz, 1]     B col n = [-2px, -2py, -2pz, |p|^2]
// score = d2 - |q|^2 (per-row shift -> identical ordering).
// Selection uses ALL 32 lanes: lane l owns query (l&15) and candidate
// columns (l>>4)*8..+7 with a private sorted top-32 list in LDS; the two
// half-lists per query are 2-pointer merged at the end. A min-tree over the
// 8 candidates gates the (rare) insert path with a single branch.
// ---------------------------------------------------------------------------
__global__ __launch_bounds__(32 * WPB) void knn_kernel(
    const float*  __restrict__ new_xyz,
    const float4* __restrict__ ptp,
    int*          __restrict__ idx_out)
{
    __shared__ float s_scores[WPB][16][SROW];
    __shared__ float s_dist[WPB][32][NS];
    __shared__ int   s_idx [WPB][32][NS];

    const int wave = threadIdx.x >> 5;
    const int lane = threadIdx.x & 31;
    const int lo   = lane & 15;
    const int hi   = lane >> 4;

    const int tile = blockIdx.x * WPB + wave;            // 0 .. BATCH*NQ/16-1
    const int b    = tile / (NQ / 16);
    const int q0   = (tile % (NQ / 16)) * 16;

    // ---- A fragment: lanes 0-15 -> (K0,K1)=(qx,qy); lanes 16-31 -> (K2,K3)=(qz,1)
    const float* qp = new_xyz + ((size_t)b * NQ + q0 + lo) * 3;
    v2f afrag;
    if (lane < 16) { afrag.x = qp[0]; afrag.y = qp[1]; }
    else           { afrag.x = qp[2]; afrag.y = 1.0f;  }

    // ---- private per-lane top-32 list (query lo, candidate half hi)
    float* mydist = &s_dist[wave][lane][0];
    int*   myidx  = &s_idx [wave][lane][0];
    for (int s = 0; s < NS; ++s) { mydist[s] = 3.4e38f; myidx[s] = 0; }
    float worst = 3.4e38f;

    const float4* pb = ptp + (size_t)b * NPTS;
    float4 p = pb[lo];                                   // tile 0, pipelined

    for (int t = 0; t < NTILES; ++t) {
        const int n0 = t * 16;

        // B fragment from the pipelined load
        v2f bfrag;
        if (lane < 16) { bfrag.x = p.x; bfrag.y = p.z; } // K0 / K2 rows
        else           { bfrag.x = p.y; bfrag.y = p.w; } // K1 / K3 rows

        v8f acc = {};
        acc = __builtin_amdgcn_wmma_f32_16x16x4_f32(
            false, afrag, false, bfrag, (short)0, acc, false, false);

        // Spill D tile to LDS (conflict-free: half-waves hit disjoint bank sets)
        #pragma unroll
        for (int r = 0; r < 8; ++r)
            s_scores[wave][r + 8 * hi][lo] = acc[r];

        // Issue next tile's load now; latency overlaps the selection scan.
        const int tn = (t + 1 < NTILES) ? (n0 + 16) : n0;
        p = pb[tn + lo];
        __builtin_prefetch(&pb[n0 + 32 + lo], 0, 1);     // global_prefetch_b8, t+2

        asm volatile("s_wait_dscnt 0x0" ::: "memory");   // scores visible wave-wide
        __builtin_amdgcn_wave_barrier();

        // Full-wave scan: two ds_load_b128 per lane cover its 8 candidates
        const float4* srow = (const float4*)&s_scores[wave][lo][hi * 8];
        float4 c0 = srow[0];
        float4 c1 = srow[1];
        float sc[8] = { c0.x, c0.y, c0.z, c0.w, c1.x, c1.y, c1.z, c1.w };

        // min-tree gate: one branch decides whether any insert can happen
        float m0 = fminf(fminf(sc[0], sc[1]), fminf(sc[2], sc[3]));
        float m1 = fminf(fminf(sc[4], sc[5]), fminf(sc[6], sc[7]));
        if (fminf(m0, m1) < worst) {                     // rare after warm-up
            const int nb = n0 + hi * 8;
            #pragma unroll
            for (int n = 0; n < 8; ++n) {
                if (sc[n] < worst) {
                    int pos = NS - 1;
                    #pragma clang loop unroll(disable)
                    while (pos > 0 && mydist[pos - 1] > sc[n]) {
                        mydist[pos] = mydist[pos - 1];
                        myidx [pos] = myidx [pos - 1];
                        --pos;
                    }
                    mydist[pos] = sc[n];
                    myidx [pos] = nb + n;
                    worst = mydist[NS - 1];
                }
            }
        }
        __builtin_amdgcn_wave_barrier();
    }

    // ---- merge the two sorted half-lists of each query, emit 32 nearest
    asm volatile("s_wait_dscnt 0x0" ::: "memory");
    __builtin_amdgcn_wave_barrier();
    if (lane < 16) {
        const float* dA = &s_dist[wave][lane     ][0];
        const int*   iA = &s_idx [wave][lane     ][0];
        const float* dB = &s_dist[wave][lane + 16][0];
        const int*   iB = &s_idx [wave][lane + 16][0];
        int* op = idx_out + ((size_t)b * NQ + q0 + lane) * NS;
        int ia = 0, ib = 0;
        #pragma clang loop unroll(disable)
        for (int s = 0; s < NS; ++s) {
            float da = dA[ia], db = dB[ib];
            if (da <= db) { op[s] = iA[ia]; ++ia; }
            else          { op[s] = iB[ib]; ++ib; }
        }
    }
}

// ---------------------------------------------------------------------------
// Kernel 2: gather. One wave per (b,q); lane = sample s.
// idx loaded once per wave (coalesced 128B) and reused for all 67 channels.
// Output stores out[b][c][q][0..31] are contiguous 128B per wave; the
// scattered 4B reads of features/xyz hit the 192MB L2 (working set 33.5MB).
// ---------------------------------------------------------------------------
__global__ __launch_bounds__(256) void gather_kernel(
    const float* __restrict__ xyz,
    const float* __restrict__ new_xyz,
    const float* __restrict__ feat,
    const int*   __restrict__ idxb,
    float*       __restrict__ out)
{
    const int wave = threadIdx.x >> 5;
    const int lane = threadIdx.x & 31;
    const int task = blockIdx.x * (blockDim.x >> 5) + wave;   // over BATCH*NQ
    if (task >= BATCH * NQ) return;
    const int b = task / NQ;
    const int q = task % NQ;

    const int n = idxb[((size_t)b * NQ + q) * NS + lane];     // lane's neighbor

    // centroid (wave-uniform -> scalarized)
    const float cx = new_xyz[((size_t)b * NQ + q) * 3 + 0];
    const float cy = new_xyz[((size_t)b * NQ + q) * 3 + 1];
    const float cz = new_xyz[((size_t)b * NQ + q) * 3 + 2];

    const float* xp = xyz + ((size_t)b * NPTS + n) * 3;
    const float gx = xp[0] - cx;
    const float gy = xp[1] - cy;
    const float gz = xp[2] - cz;

    const size_t cstride = (size_t)NQ * NS;                       // channel stride
    float* ob = out + (((size_t)b * COUT) * NQ + q) * NS + lane;  // c = 0 slot
    ob[0 * cstride] = gx;
    ob[1 * cstride] = gy;
    ob[2 * cstride] = gz;

    const float* fb = feat + ((size_t)b * NC) * NPTS + n;
    #pragma unroll 4
    for (int c = 0; c < NC; ++c)
        ob[(size_t)(3 + c) * cstride] = fb[(size_t)c * NPTS];
}

// ---------------------------------------------------------------------------
extern "C" void kernel_launch(void* const* d_in, const int* in_sizes, int n_in,
                              void* d_out, int out_size, void* d_ws, size_t ws_size,
                              hipStream_t stream)
{
    const float* xyz     = (const float*)d_in[0];   // (8,16384,3) f32
    const float* new_xyz = (const float*)d_in[1];   // (8,2048,3)  f32
    const float* feat    = (const float*)d_in[2];   // (8,64,16384) f32
    float* out = (float*)d_out;                     // (8,67,2048,32) f32

    // workspace: [0, 2MB) packed points, [2MB, 4MB) neighbor indices
    float4* ptp  = (float4*)d_ws;
    int*    idxb = (int*)((char*)d_ws + (size_t)BATCH * NPTS * sizeof(float4));

    pack_points_kernel<<<(BATCH * NPTS + 255) / 256, 256, 0, stream>>>(xyz, ptp);

    const int tiles = BATCH * NQ / 16;              // 1024 query tiles
    knn_kernel<<<tiles / WPB, 32 * WPB, 0, stream>>>(new_xyz, ptp, idxb);

    gather_kernel<<<(BATCH * NQ) / 8, 256, 0, stream>>>(xyz, new_xyz, feat, idxb, out);
}